// Encoder_89180700934746
// MI455X (gfx1250) — compile-verified
//
#include <hip/hip_runtime.h>
#include <hip/hip_bf16.h>
#include <limits.h>

// ---------------------------------------------------------------------------
// GAT 2-layer forward for MI455X (gfx1250, wave32).
//  - GEMM h = x @ W via V_WMMA_F32_16X16X4_F32 (fp32 WMMA, keeps ref numerics)
//  - edge phase via float atomics (ordered-int atomicMax for segment max,
//    global_atomic_add_f32 for segment sum / scatter)
// ---------------------------------------------------------------------------

typedef float v2f __attribute__((ext_vector_type(2)));
typedef float v8f __attribute__((ext_vector_type(8)));

#define DFEAT 64
#define NEG_SLOPE 0.2f
#define EPS_GAT 1e-16f

// ---- float <-> order-preserving int keys (exact float max via atomicMax) ---
__device__ __forceinline__ int f2ord(float f) {
    int i = __float_as_int(f);
    return i >= 0 ? i : (i ^ 0x7FFFFFFF);
}
__device__ __forceinline__ float ord2f(int i) {
    int j = i >= 0 ? i : (i ^ 0x7FFFFFFF);
    return __int_as_float(j);
}

// ---------------------------------------------------------------------------
// out[n][d] = b[d]; mkey[n] = -inf key; s[n] = 0
// ---------------------------------------------------------------------------
__global__ void __launch_bounds__(256)
k_init(float* __restrict__ out, const float* __restrict__ b,
       int* __restrict__ mkey, float* __restrict__ s, int N) {
    int i = blockIdx.x * 256 + threadIdx.x;
    int total = N * DFEAT;
    if (i < total) out[i] = b[i & (DFEAT - 1)];
    if (i < N) { mkey[i] = INT_MIN; s[i] = 0.0f; }
}

// ---------------------------------------------------------------------------
// h = x @ W   (x: [N,64] f32, W: [64,64] f32, h: [N,64] f32)
// 256 threads = 8 waves; each wave owns a 16-row tile (block covers 128 rows).
// K=64 -> 16 WMMA f32 16x16x4 steps per 16x16 output tile, 4 N-tiles.
//
// fp32 WMMA VGPR layouts (ISA 7.12.2):
//   A 16x4:  lanes 0-15 hold M=lane, VGPR0=K0,VGPR1=K1; lanes 16-31: K2,K3
//   B 4x16:  row-striped across lanes: VGPR0 lanes0-15 K=0 / lanes16-31 K=2,
//            VGPR1 K=1 / K=3  (N = lane&15)
//   C/D 16x16: VGPR j -> M = j + (lane>=16)*8, N = lane&15
// ---------------------------------------------------------------------------
__global__ void __launch_bounds__(256)
k_gemm(const float* __restrict__ x, const float* __restrict__ W,
       float* __restrict__ h, int N) {
    __shared__ float Ws[DFEAT * DFEAT];     // 16 KB of the 320 KB WGP LDS
    for (int i = threadIdx.x; i < DFEAT * DFEAT; i += 256) Ws[i] = W[i];
    __syncthreads();

    const int wave  = threadIdx.x >> 5;
    const int lane  = threadIdx.x & 31;
    const int m0    = (blockIdx.x * 8 + wave) * 16;
    if (m0 >= N) return;

    const int mrow  = lane & 15;
    const int khalf = lane >> 4;            // 0 or 1
    int gm = m0 + mrow; if (gm >= N) gm = N - 1;
    const float* __restrict__ xrow = x + (size_t)gm * DFEAT;

    const int ncol_in_tile = lane & 15;

    for (int nt = 0; nt < 4; ++nt) {
        v8f acc = {};
#pragma unroll
        for (int k = 0; k < DFEAT; k += 4) {
            v2f a, bfrag;
            a.x     = xrow[k + khalf * 2 + 0];
            a.y     = xrow[k + khalf * 2 + 1];
            bfrag.x = Ws[(k + khalf * 2 + 0) * DFEAT + nt * 16 + ncol_in_tile];
            bfrag.y = Ws[(k + khalf * 2 + 1) * DFEAT + nt * 16 + ncol_in_tile];
            acc = __builtin_amdgcn_wmma_f32_16x16x4_f32(
                /*neg_a=*/false, a, /*neg_b=*/false, bfrag,
                /*c_mod=*/(short)0, acc, /*reuse_a=*/false, /*reuse_b=*/false);
        }
        const int ncol = nt * 16 + ncol_in_tile;
#pragma unroll
        for (int j = 0; j < 8; ++j) {
            int m = m0 + j + khalf * 8;
            if (m < N) h[(size_t)m * DFEAT + ncol] = acc[j];
        }
    }
}

// ---------------------------------------------------------------------------
// alpha_src[n] = h[n]·a_src ; alpha_dst[n] = h[n]·a_dst  (wave per node)
// ---------------------------------------------------------------------------
__global__ void __launch_bounds__(256)
k_alpha(const float* __restrict__ h, const float* __restrict__ a_src,
        const float* __restrict__ a_dst, float* __restrict__ asr,
        float* __restrict__ ads, int N) {
    int wave = threadIdx.x >> 5, lane = threadIdx.x & 31;
    int n = blockIdx.x * 8 + wave;
    if (n >= N) return;
    const float* hr = h + (size_t)n * DFEAT;
    float v0 = hr[lane], v1 = hr[lane + 32];
    float ps = v0 * a_src[lane] + v1 * a_src[lane + 32];
    float pd = v0 * a_dst[lane] + v1 * a_dst[lane + 32];
#pragma unroll
    for (int off = 16; off > 0; off >>= 1) {
        ps += __shfl_xor(ps, off, 32);
        pd += __shfl_xor(pd, off, 32);
    }
    if (lane == 0) { asr[n] = ps; ads[n] = pd; }
}

// ---------------------------------------------------------------------------
// Pass 1 over edges: e = leaky_relu(asrc[src]+adst[dst]); segment max by dst.
// ---------------------------------------------------------------------------
__global__ void __launch_bounds__(256)
k_edge_max(const long long* __restrict__ srcp, const long long* __restrict__ dstp,
           int nseg, const float* __restrict__ asr, const float* __restrict__ ads,
           int* __restrict__ mkey, float* __restrict__ eb) {
    int i = blockIdx.x * 256 + threadIdx.x;
    if (i >= nseg) return;
    int src = (int)srcp[i];
    int dst = (int)dstp[i];
    float v = asr[src] + ads[dst];
    v = (v >= 0.0f) ? v : NEG_SLOPE * v;
    eb[i] = v;
    atomicMax(&mkey[dst], f2ord(v));
}

// ---------------------------------------------------------------------------
// Pass 2: ex = exp(e - max[dst]); segment sum by dst.
// ---------------------------------------------------------------------------
__global__ void __launch_bounds__(256)
k_edge_expsum(const long long* __restrict__ dstp, int nseg,
              const int* __restrict__ mkey, float* __restrict__ s,
              float* __restrict__ eb) {
    int i = blockIdx.x * 256 + threadIdx.x;
    if (i >= nseg) return;
    int dst = (int)dstp[i];
    float ex = __expf(eb[i] - ord2f(mkey[dst]));
    eb[i] = ex;
    atomicAdd(&s[dst], ex);
}

// ---------------------------------------------------------------------------
// Pass 3: alpha = ex/(s[dst]+eps); out[dst] += alpha * h[src].
// 64 threads per edge (one feature each), 4 edges per 256-thread block.
// ---------------------------------------------------------------------------
__global__ void __launch_bounds__(256)
k_edge_scatter(const long long* __restrict__ srcp, const long long* __restrict__ dstp,
               int nseg, const float* __restrict__ h, const float* __restrict__ s,
               const float* __restrict__ eb, float* __restrict__ out) {
    int i = blockIdx.x * 4 + (threadIdx.x >> 6);
    if (i >= nseg) return;
    int t   = threadIdx.x & 63;
    int src = (int)srcp[i];
    int dst = (int)dstp[i];
    float alpha = eb[i] / (s[dst] + EPS_GAT);
    atomicAdd(&out[(size_t)dst * DFEAT + t], h[(size_t)src * DFEAT + t] * alpha);
}

// ---------------------------------------------------------------------------
static void run_gat_layer(const float* xin, const float* W, const float* a_s,
                          const float* a_d, const float* b,
                          const long long* src0, const long long* dst0,
                          long long e0, long long e1, float* out,
                          float* h, float* asr, float* ads,
                          int* mkey, float* s, float* eb, int N,
                          hipStream_t stream) {
    int nseg = (int)(e1 - e0);
    k_init<<<(N * DFEAT + 255) / 256, 256, 0, stream>>>(out, b, mkey, s, N);
    k_gemm<<<(N + 127) / 128, 256, 0, stream>>>(xin, W, h, N);
    k_alpha<<<(N + 7) / 8, 256, 0, stream>>>(h, a_s, a_d, asr, ads, N);
    if (nseg > 0) {
        k_edge_max<<<(nseg + 255) / 256, 256, 0, stream>>>(
            src0 + e0, dst0 + e0, nseg, asr, ads, mkey, eb);
        k_edge_expsum<<<(nseg + 255) / 256, 256, 0, stream>>>(
            dst0 + e0, nseg, mkey, s, eb);
        k_edge_scatter<<<(nseg + 3) / 4, 256, 0, stream>>>(
            src0 + e0, dst0 + e0, nseg, h, s, eb, out);
    }
}

extern "C" void kernel_launch(void* const* d_in, const int* in_sizes, int n_in,
                              void* d_out, int out_size, void* d_ws, size_t ws_size,
                              hipStream_t stream) {
    const float*     x      = (const float*)d_in[0];
    const long long* ei     = (const long long*)d_in[1];   // [2, E] int64
    const float*     W1     = (const float*)d_in[2];
    const float*     a_src1 = (const float*)d_in[3];
    const float*     a_dst1 = (const float*)d_in[4];
    const float*     b1     = (const float*)d_in[5];
    const float*     W2     = (const float*)d_in[6];
    const float*     a_src2 = (const float*)d_in[7];
    const float*     a_dst2 = (const float*)d_in[8];
    const float*     b2     = (const float*)d_in[9];

    const int       N = in_sizes[0] / DFEAT;
    const long long E = in_sizes[1] / 2;
    const long long ESPLIT = (E < 500) ? E : 500;

    const long long* src0 = ei;        // row 0
    const long long* dst0 = ei + E;    // row 1

    // workspace carve-up (floats): h | x1 | asrc | adst | s | mkey | ebuf
    float* ws  = (float*)d_ws;
    float* h   = ws;
    float* x1  = h  + (size_t)N * DFEAT;
    float* asr = x1 + (size_t)N * DFEAT;
    float* ads = asr + N;
    float* s   = ads + N;
    int*   mk  = (int*)(s + N);
    float* eb  = (float*)(mk + N);

    float* out = (float*)d_out;

    // Layer 1: conv_near on edges [0, 500)  -> x1
    run_gat_layer(x,  W1, a_src1, a_dst1, b1, src0, dst0, 0, ESPLIT,
                  x1, h, asr, ads, mk, s, eb, N, stream);
    // Layer 2: conv_include on edges [500, E) -> d_out
    run_gat_layer(x1, W2, a_src2, a_dst2, b2, src0, dst0, ESPLIT, E,
                  out, h, asr, ads, mk, s, eb, N, stream);
    (void)n_in; (void)out_size; (void)ws_size;
}